// OnlineTripletLoss_61108794688119
// MI455X (gfx1250) — compile-verified
//
#include <hip/hip_runtime.h>

// ---------------------------------------------------------------------------
// Batch-hard online triplet loss for MI455X (gfx1250, wave32).
//   X: [8192, 64] f32, targets: [8192] int
//   dist[i][j] = max(sq[i] + sq[j] - 2*dot(x_i,x_j), 0)
//   hard_pos[i] = first argmax over same-class, hard_neg[i] = first argmin
//   over other-class; loss = mean(relu(ap - an + 0.5))
//
// Gram tiles via V_WMMA_F32_16X16X4_F32 (fp32 matrix pipe = reference
// precision). 2 MB working set lives in L2 (192 MB): compute bound.
// Round-2 changes: LDS row stride padded 64->68 floats (conflict-free B
// loads), B fragments batch-loaded + double-buffered across tiles, column
// dimension split into 4 slices (512 WGs) with a deterministic merge.
// ---------------------------------------------------------------------------

typedef float v2f __attribute__((ext_vector_type(2)));
typedef float v8f __attribute__((ext_vector_type(8)));

#define NROWS 8192
#define DIM   64
#define NCHUNK 128                     // columns staged in LDS per chunk
#define LDS_STRIDE 68                  // 64 + 4 pad floats: banks = 4c + k
#define NSLICES 4
#define COLS_PER_SLICE (NROWS / NSLICES)   // 2048
#define WAVES_PER_WG 4
#define WG_THREADS (WAVES_PER_WG * 32)
#define ROWS_PER_WAVE 16
#define ROWS_PER_WG (WAVES_PER_WG * ROWS_PER_WAVE)   // 64
#define MARGIN 0.5f

// ---------------------------------------------------------------- sq norms --
__global__ __launch_bounds__(256)
void sq_kernel(const float* __restrict__ X, float* __restrict__ sq) {
    int i = blockIdx.x * blockDim.x + threadIdx.x;
    if (i >= NROWS) return;
    const float4* r = (const float4*)(X + i * DIM);
    float s = 0.0f;
#pragma unroll
    for (int t = 0; t < DIM / 4; ++t) {
        float4 v = r[t];
        s += v.x * v.x + v.y * v.y + v.z * v.z + v.w * v.w;
    }
    sq[i] = s;
}

// ------------------------------------------------------------ hard mining --
// Block (bx, slice): 64 anchor rows vs columns [slice*2048, slice*2048+2048).
// Wave handles 16 anchor rows. WMMA f32 16x16x4 fragment layout (ISA 7.12.2):
//   lane l: index (l&15), K-pair offset 2*(l>>4); A (M-major) and B (N-major)
//   fragments of X@X^T have the identical per-lane pattern.
// C/D: VGPR v holds row m = v + 8*(lane>>4), column n = lane&15.
__global__ __launch_bounds__(WG_THREADS)
void mine_kernel(const float* __restrict__ X, const float* __restrict__ sq,
                 const int* __restrict__ tgt,
                 float* __restrict__ posVal, int* __restrict__ posIdx,
                 float* __restrict__ negVal, int* __restrict__ negIdx) {
    __shared__ __align__(16) float sX[NCHUNK * LDS_STRIDE];  // ~34 KB
    __shared__ float sSq[NCHUNK];
    __shared__ int   sT[NCHUNK];

    const int tid   = threadIdx.x;
    const int wave  = tid >> 5;
    const int lane  = tid & 31;
    const int nIdx  = lane & 15;
    const int half  = lane >> 4;
    const int rowBase  = blockIdx.x * ROWS_PER_WG + wave * ROWS_PER_WAVE;
    const int slice    = blockIdx.y;
    const int colStart = slice * COLS_PER_SLICE;

    // A fragments for all 16 K-chunks: resident in VGPRs for the whole kernel.
    v2f afrag[DIM / 4];
    const float* arow = X + (rowBase + nIdx) * DIM;
#pragma unroll
    for (int kc = 0; kc < DIM / 4; ++kc) {
        int k = kc * 4 + 2 * half;
        afrag[kc].x = arow[k];
        afrag[kc].y = arow[k + 1];
    }

    float sqa[8];
    int   ta[8];
#pragma unroll
    for (int v = 0; v < 8; ++v) {
        int m = v + 8 * half;
        sqa[v] = sq[rowBase + m];
        ta[v]  = tgt[rowBase + m];
    }

    float posB[8], negB[8];
    int   posI[8], negI[8];
#pragma unroll
    for (int v = 0; v < 8; ++v) {
        posB[v] = -__builtin_inff();
        negB[v] =  __builtin_inff();
        posI[v] = 0;
        negI[v] = 0;
    }

    // B-fragment loader: 16x ds_load pairs per tile, conflict-free banks.
    auto loadB = [&](int tile, v2f* bf) {
        const float* bcol = sX + (tile * 16 + nIdx) * LDS_STRIDE;
#pragma unroll
        for (int kc = 0; kc < DIM / 4; ++kc) {
            int k = kc * 4 + 2 * half;
            bf[kc].x = bcol[k];
            bf[kc].y = bcol[k + 1];
        }
    };

    for (int cb = 0; cb < COLS_PER_SLICE; cb += NCHUNK) {
        __syncthreads();
        // Stage 128 columns: one column per thread, padded stride in LDS.
        {
            const float4* src = (const float4*)(X + (colStart + cb + tid) * DIM);
            float4* dst = (float4*)(sX + tid * LDS_STRIDE);
#pragma unroll
            for (int t = 0; t < DIM / 4; ++t) dst[t] = src[t];
            sSq[tid] = sq[colStart + cb + tid];
            sT[tid]  = tgt[colStart + cb + tid];
        }
        __syncthreads();

        // Warm L2 for the next chunk (global_prefetch_b8), 256B per thread.
        if (cb + NCHUNK < COLS_PER_SLICE) {
            const float* nxt = X + (colStart + cb + NCHUNK) * DIM;
            __builtin_prefetch(nxt + tid * 64, 0, 1);
            __builtin_prefetch(nxt + tid * 64 + 32, 0, 1);
        }

        v2f bufA[DIM / 4], bufB[DIM / 4];
        loadB(0, bufA);
#pragma unroll
        for (int tile = 0; tile < NCHUNK / 16; ++tile) {
            v2f* cur = (tile & 1) ? bufB : bufA;
            v2f* nxt = (tile & 1) ? bufA : bufB;
            if (tile + 1 < NCHUNK / 16) loadB(tile + 1, nxt);  // pipeline ahead

            v8f acc = {0.f, 0.f, 0.f, 0.f, 0.f, 0.f, 0.f, 0.f};
#pragma unroll
            for (int kc = 0; kc < DIM / 4; ++kc)
                acc = __builtin_amdgcn_wmma_f32_16x16x4_f32(
                    false, afrag[kc], false, cur[kc], (short)0, acc, false, false);

            const int   j   = colStart + cb + tile * 16 + nIdx;
            const float sqb = sSq[tile * 16 + nIdx];
            const int   tb  = sT[tile * 16 + nIdx];
#pragma unroll
            for (int v = 0; v < 8; ++v) {
                float d = fmaxf(sqa[v] + sqb - 2.0f * acc[v], 0.0f);
                bool same = (ta[v] == tb);
                float pc = same ? d : -__builtin_inff();
                float nc = same ?  __builtin_inff() : d;
                // strict updates => earliest index per lane (first occurrence)
                if (pc > posB[v]) { posB[v] = pc; posI[v] = j; }
                if (nc < negB[v]) { negB[v] = nc; negI[v] = j; }
            }
        }
    }

    // Cross-lane reduce within each 16-lane group: value first, lower index
    // on ties => global first-occurrence (each column owned by one lane).
#pragma unroll
    for (int v = 0; v < 8; ++v) {
        float pb = posB[v]; int pi = posI[v];
        float nb = negB[v]; int ni = negI[v];
#pragma unroll
        for (int off = 1; off < 16; off <<= 1) {
            float opb = __shfl_xor(pb, off, 32);
            int   opi = __shfl_xor(pi, off, 32);
            if (opb > pb || (opb == pb && opi < pi)) { pb = opb; pi = opi; }
            float onb = __shfl_xor(nb, off, 32);
            int   oni = __shfl_xor(ni, off, 32);
            if (onb < nb || (onb == nb && oni < ni)) { nb = onb; ni = oni; }
        }
        if (nIdx == v) {
            int row = rowBase + v + 8 * half;
            posVal[slice * NROWS + row] = pb;
            posIdx[slice * NROWS + row] = pi;
            negVal[slice * NROWS + row] = nb;
            negIdx[slice * NROWS + row] = ni;
        }
    }
}

// --------------------------------------- slice merge + loss + block reduce --
__global__ __launch_bounds__(256)
void loss_kernel(const float* __restrict__ X,
                 const float* __restrict__ posVal, const int* __restrict__ posIdx,
                 const float* __restrict__ negVal, const int* __restrict__ negIdx,
                 float* __restrict__ partial) {
    int i = blockIdx.x * blockDim.x + threadIdx.x;

    // Merge slices in ascending order; strict compares keep the earliest
    // index among maxima/minima (slices cover ascending column ranges).
    float pb = -__builtin_inff(); int pi = 0;
    float nb =  __builtin_inff(); int ni = 0;
#pragma unroll
    for (int s = 0; s < NSLICES; ++s) {
        float v = posVal[s * NROWS + i]; int id = posIdx[s * NROWS + i];
        if (v > pb) { pb = v; pi = id; }
        float w = negVal[s * NROWS + i]; int jd = negIdx[s * NROWS + i];
        if (w < nb) { nb = w; ni = jd; }
    }

    // Recompute exact squared distances from embedding diffs (as reference).
    float ap = 0.0f, an = 0.0f;
    const float* xi = X + i * DIM;
    const float* xp = X + pi * DIM;
    const float* xn = X + ni * DIM;
#pragma unroll
    for (int d = 0; d < DIM; ++d) {
        float dp = xi[d] - xp[d]; ap += dp * dp;
        float dn = xi[d] - xn[d]; an += dn * dn;
    }
    float loss = fmaxf(ap - an + MARGIN, 0.0f);

    __shared__ float red[256];
    red[threadIdx.x] = loss;
    __syncthreads();
#pragma unroll
    for (int s = 128; s > 0; s >>= 1) {
        if (threadIdx.x < s) red[threadIdx.x] += red[threadIdx.x + s];
        __syncthreads();
    }
    if (threadIdx.x == 0) partial[blockIdx.x] = red[0];
}

__global__ void final_kernel(const float* __restrict__ partial,
                             float* __restrict__ out, int nb) {
    if (threadIdx.x == 0 && blockIdx.x == 0) {
        float s = 0.0f;
        for (int b = 0; b < nb; ++b) s += partial[b];   // fixed order: deterministic
        out[0] = s / (float)NROWS;
    }
}

// --------------------------------------------------------------- launch ---
extern "C" void kernel_launch(void* const* d_in, const int* in_sizes, int n_in,
                              void* d_out, int out_size, void* d_ws, size_t ws_size,
                              hipStream_t stream) {
    (void)in_sizes; (void)n_in; (void)out_size; (void)ws_size;
    const float* X   = (const float*)d_in[0];
    const int*   tgt = (const int*)d_in[1];
    float* out = (float*)d_out;

    char* ws = (char*)d_ws;
    size_t off = 0;
    float* posVal = (float*)(ws + off); off += (size_t)NSLICES * NROWS * 4;
    int*   posIdx = (int*)  (ws + off); off += (size_t)NSLICES * NROWS * 4;
    float* negVal = (float*)(ws + off); off += (size_t)NSLICES * NROWS * 4;
    int*   negIdx = (int*)  (ws + off); off += (size_t)NSLICES * NROWS * 4;
    float* sq      = (float*)(ws + off); off += (size_t)NROWS * 4;
    float* partial = (float*)(ws + off);

    sq_kernel<<<NROWS / 256, 256, 0, stream>>>(X, sq);
    dim3 grid(NROWS / ROWS_PER_WG, NSLICES);
    mine_kernel<<<grid, WG_THREADS, 0, stream>>>(X, sq, tgt,
                                                 posVal, posIdx, negVal, negIdx);
    loss_kernel<<<NROWS / 256, 256, 0, stream>>>(X, posVal, posIdx,
                                                 negVal, negIdx, partial);
    final_kernel<<<1, 32, 0, stream>>>(partial, out, NROWS / 256);
}